// EGNN_static_24395414242137
// MI455X (gfx1250) — compile-verified
//
#include <hip/hip_runtime.h>

typedef float v8f __attribute__((ext_vector_type(8)));
typedef float v4f __attribute__((ext_vector_type(4)));
typedef float v2f __attribute__((ext_vector_type(2)));

#define N_NODES 50000
#define N_EDGES 800000
#define D 64
#define WPB 8                 // waves per block
#define BLOCK (WPB * 32)
#define E1_STRIDE 68          // 16x64 tile padded: bank = (4m+k)&63, conflict-free

// d_ws float layout:
//   [0,     8192)  swizzled w1e B-fragments (32 kc * 2 ntpair * 32 lanes * 4)
//   [8192, 12288)  swizzled w2e (16 kc)
//   [12288,20480)  swizzled w1n (32 kc)
//   [20480,24576)  swizzled w2n (16 kc)
//   [24576, 24576 + N*64)  agg accumulator
#define WS_W1E 0
#define WS_W2E 8192
#define WS_W1N 12288
#define WS_W2N 20480
#define WS_AGG 24576

__device__ __forceinline__ float lrelu(float x) { return fmaxf(x, 0.2f * x); }

// Pre-swizzle a [K x 64] weight into b128-fragment-pair order:
// entry idx -> (kc, ntpair, lane); the 4 floats are
//   { frag(nt=2p)[j=0], frag(2p)[j=1], frag(2p+1)[j=0], frag(2p+1)[j=1] }
// where fragment element j for lane l is w[kc*4 + (l>>4)*2 + j][nt*16 + (l&15)].
__device__ __forceinline__ void swz_one(const float* __restrict__ w,
                                        float* __restrict__ dst, int idx) {
    int lane = idx & 31;
    int ntp  = (idx >> 5) & 1;
    int kc   = idx >> 6;
    int k  = kc * 4 + ((lane >> 4) << 1);
    int n0 = ntp * 32 + (lane & 15);        // nt = 2p
    int n1 = n0 + 16;                       // nt = 2p+1
    dst[idx * 4 + 0] = w[(size_t)k * D + n0];
    dst[idx * 4 + 1] = w[(size_t)(k + 1) * D + n0];
    dst[idx * 4 + 2] = w[(size_t)k * D + n1];
    dst[idx * 4 + 3] = w[(size_t)(k + 1) * D + n1];
}

__global__ void egnn_swizzle_kernel(const float* __restrict__ w1e,
                                    const float* __restrict__ w2e,
                                    const float* __restrict__ w1n,
                                    const float* __restrict__ w2n,
                                    float* __restrict__ ws) {
    int t = threadIdx.x;
    for (int i = t; i < 32 * 2 * 32; i += 256) swz_one(w1e, ws + WS_W1E, i);
    for (int i = t; i < 16 * 2 * 32; i += 256) swz_one(w2e, ws + WS_W2E, i);
    for (int i = t; i < 32 * 2 * 32; i += 256) swz_one(w1n, ws + WS_W1N, i);
    for (int i = t; i < 16 * 2 * 32; i += 256) swz_one(w2n, ws + WS_W2N, i);
}

__global__ void egnn_zero_kernel(float* __restrict__ p, int n) {
    int i = blockIdx.x * blockDim.x + threadIdx.x;
    if (i < n) p[i] = 0.0f;
}

// 4 WMMAs of one K-chunk against accumulators
__device__ __forceinline__ void wmma4(v8f acc[4], v2f a, v4f b01, v4f b23) {
    acc[0] = __builtin_amdgcn_wmma_f32_16x16x4_f32(false, a, false, b01.xy,
                                                   (short)0, acc[0], false, false);
    acc[1] = __builtin_amdgcn_wmma_f32_16x16x4_f32(false, a, false, b01.zw,
                                                   (short)0, acc[1], false, false);
    acc[2] = __builtin_amdgcn_wmma_f32_16x16x4_f32(false, a, false, b23.xy,
                                                   (short)0, acc[2], false, false);
    acc[3] = __builtin_amdgcn_wmma_f32_16x16x4_f32(false, a, false, b23.zw,
                                                   (short)0, acc[3], false, false);
}

// ---------------- Edge kernel: edge MLP + scatter-add --------------------
__global__ __launch_bounds__(BLOCK)
void egnn_edge_kernel(const float* __restrict__ h,
                      const float* __restrict__ coord,
                      const int*   __restrict__ ei,
                      const float* __restrict__ w1e_raw,  // [129 x 64]; row 128 = radial weights
                      const float* __restrict__ b1e,
                      const float* __restrict__ b2e,
                      const float* __restrict__ sw1,
                      const float* __restrict__ sw2,
                      float* __restrict__ agg,
                      float* __restrict__ ef_out) {
    __shared__ __align__(16) float sE1[WPB][16 * E1_STRIDE];

    const int lane = threadIdx.x & 31;
    const int wave = threadIdx.x >> 5;
    const int tile = blockIdx.x * WPB + wave;
    if (tile * 16 >= N_EDGES) return;

    const int m    = lane & 15;          // edge-in-tile for A fragments
    const int half = lane >> 4;
    const int kk2  = half << 1;          // K sub-offset within chunk: 0 or 2
    const int n0   = lane & 15;          // column within an N-tile

    const int row = ei[tile * 16 + m];
    const int col = ei[N_EDGES + tile * 16 + m];

    float dx = coord[row * 3 + 0] - coord[col * 3 + 0];
    float dy = coord[row * 3 + 1] - coord[col * 3 + 1];
    float dz = coord[row * 3 + 2] - coord[col * 3 + 2];
    float radial = dx * dx + dy * dy + dz * dz;

    // per-accumulator-row (m = r + half*8) values of radial and row index
    float radm[8];
    int   rowm[8];
#pragma unroll
    for (int r = 0; r < 8; ++r) {
        int src = r + (half << 3);
        radm[r] = __shfl(radial, src, 32);
        rowm[r] = __shfl(row, src, 32);
    }

    // C init: b1e[n] + radial[m] * w1e[128][n]   (folds the K=129th column)
    const float* wrad = w1e_raw + 128 * D;
    v8f acc[4];
#pragma unroll
    for (int nt = 0; nt < 4; ++nt) {
        float b  = b1e[nt * 16 + n0];
        float wr = wrad[nt * 16 + n0];
#pragma unroll
        for (int r = 0; r < 8; ++r) acc[nt][r] = fmaf(radm[r], wr, b);
    }

    const float* aRow = h + (size_t)row * D + kk2;
    const float* aCol = h + (size_t)col * D + kk2;

    // ---- Layer 1: K = 128 (h[row] || h[col]), software-pipelined ----
    v2f a   = *(const v2f*)(aRow);
    v4f b01 = *(const v4f*)(sw1 + (size_t)((0 * 2 + 0) * 32 + lane) * 4);
    v4f b23 = *(const v4f*)(sw1 + (size_t)((0 * 2 + 1) * 32 + lane) * 4);
#pragma unroll
    for (int kc = 0; kc < 32; ++kc) {
        const int kn = (kc + 1 < 32) ? kc + 1 : 31;   // prefetch index
        v2f a_n = (kn < 16) ? *(const v2f*)(aRow + kn * 4)
                            : *(const v2f*)(aCol + (kn - 16) * 4);
        v4f b01_n = *(const v4f*)(sw1 + (size_t)((kn * 2 + 0) * 32 + lane) * 4);
        v4f b23_n = *(const v4f*)(sw1 + (size_t)((kn * 2 + 1) * 32 + lane) * 4);
        wmma4(acc, a, b01, b23);
        a = a_n; b01 = b01_n; b23 = b23_n;
    }

    // lrelu + D-layout -> LDS (row-major 16x64, stride 68)
    float* e1 = sE1[wave];
#pragma unroll
    for (int nt = 0; nt < 4; ++nt) {
#pragma unroll
        for (int r = 0; r < 8; ++r) {
            int mm = r + (half << 3);
            e1[mm * E1_STRIDE + nt * 16 + n0] = lrelu(acc[nt][r]);
        }
    }
    asm volatile("" ::: "memory");  // wave-synchronous; keep compiler ordered

    // ---- Layer 2: K = 64, A from LDS (transpose), pipelined ----
    v8f acc2[4];
#pragma unroll
    for (int nt = 0; nt < 4; ++nt) {
        float b = b2e[nt * 16 + n0];
#pragma unroll
        for (int r = 0; r < 8; ++r) acc2[nt][r] = b;
    }
    const float* aL = e1 + m * E1_STRIDE + kk2;
    a   = *(const v2f*)(aL);
    b01 = *(const v4f*)(sw2 + (size_t)((0 * 2 + 0) * 32 + lane) * 4);
    b23 = *(const v4f*)(sw2 + (size_t)((0 * 2 + 1) * 32 + lane) * 4);
#pragma unroll
    for (int kc = 0; kc < 16; ++kc) {
        const int kn = (kc + 1 < 16) ? kc + 1 : 15;
        v2f a_n   = *(const v2f*)(aL + kn * 4);
        v4f b01_n = *(const v4f*)(sw2 + (size_t)((kn * 2 + 0) * 32 + lane) * 4);
        v4f b23_n = *(const v4f*)(sw2 + (size_t)((kn * 2 + 1) * 32 + lane) * 4);
        wmma4(acc2, a, b01, b23);
        a = a_n; b01 = b01_n; b23 = b23_n;
    }

    // lrelu, write edge_feat, scatter-add into agg
#pragma unroll
    for (int r = 0; r < 8; ++r) {
        int mm = r + (half << 3);
        size_t erow = (size_t)(tile * 16 + mm) * D;
        size_t arow = (size_t)rowm[r] * D;
#pragma unroll
        for (int nt = 0; nt < 4; ++nt) {
            float v = lrelu(acc2[nt][r]);
            ef_out[erow + nt * 16 + n0] = v;
            atomicAdd(agg + arow + nt * 16 + n0, v);
        }
    }
}

// ---------------- Node kernel: node MLP + residual -----------------------
__global__ __launch_bounds__(BLOCK)
void egnn_node_kernel(const float* __restrict__ h,
                      const float* __restrict__ agg,
                      const float* __restrict__ b1n,
                      const float* __restrict__ b2n,
                      const float* __restrict__ sw1,
                      const float* __restrict__ sw2,
                      float* __restrict__ h_out) {
    __shared__ __align__(16) float sE1[WPB][16 * E1_STRIDE];

    const int lane = threadIdx.x & 31;
    const int wave = threadIdx.x >> 5;
    const int tile = blockIdx.x * WPB + wave;
    if (tile * 16 >= N_NODES) return;

    const int m    = lane & 15;
    const int half = lane >> 4;
    const int kk2  = half << 1;
    const int n0   = lane & 15;

    const float* aRow = h   + (size_t)(tile * 16 + m) * D + kk2;
    const float* aAgg = agg + (size_t)(tile * 16 + m) * D + kk2;

    v8f acc[4];
#pragma unroll
    for (int nt = 0; nt < 4; ++nt) {
        float b = b1n[nt * 16 + n0];
#pragma unroll
        for (int r = 0; r < 8; ++r) acc[nt][r] = b;
    }

    // ---- Layer 1: K = 128 (h[node] || agg[node]), pipelined ----
    v2f a   = *(const v2f*)(aRow);
    v4f b01 = *(const v4f*)(sw1 + (size_t)((0 * 2 + 0) * 32 + lane) * 4);
    v4f b23 = *(const v4f*)(sw1 + (size_t)((0 * 2 + 1) * 32 + lane) * 4);
#pragma unroll
    for (int kc = 0; kc < 32; ++kc) {
        const int kn = (kc + 1 < 32) ? kc + 1 : 31;
        v2f a_n = (kn < 16) ? *(const v2f*)(aRow + kn * 4)
                            : *(const v2f*)(aAgg + (kn - 16) * 4);
        v4f b01_n = *(const v4f*)(sw1 + (size_t)((kn * 2 + 0) * 32 + lane) * 4);
        v4f b23_n = *(const v4f*)(sw1 + (size_t)((kn * 2 + 1) * 32 + lane) * 4);
        wmma4(acc, a, b01, b23);
        a = a_n; b01 = b01_n; b23 = b23_n;
    }

    float* e1 = sE1[wave];
#pragma unroll
    for (int nt = 0; nt < 4; ++nt) {
#pragma unroll
        for (int r = 0; r < 8; ++r) {
            int mm = r + (half << 3);
            e1[mm * E1_STRIDE + nt * 16 + n0] = lrelu(acc[nt][r]);
        }
    }
    asm volatile("" ::: "memory");

    // ---- Layer 2: K = 64; C init = b2n[n] + h[node][n] (residual) ----
    v8f acc2[4];
#pragma unroll
    for (int nt = 0; nt < 4; ++nt) {
        float b = b2n[nt * 16 + n0];
#pragma unroll
        for (int r = 0; r < 8; ++r) {
            int mm = r + (half << 3);
            acc2[nt][r] = b + h[(size_t)(tile * 16 + mm) * D + nt * 16 + n0];
        }
    }
    const float* aL = e1 + m * E1_STRIDE + kk2;
    a   = *(const v2f*)(aL);
    b01 = *(const v4f*)(sw2 + (size_t)((0 * 2 + 0) * 32 + lane) * 4);
    b23 = *(const v4f*)(sw2 + (size_t)((0 * 2 + 1) * 32 + lane) * 4);
#pragma unroll
    for (int kc = 0; kc < 16; ++kc) {
        const int kn = (kc + 1 < 16) ? kc + 1 : 15;
        v2f a_n   = *(const v2f*)(aL + kn * 4);
        v4f b01_n = *(const v4f*)(sw2 + (size_t)((kn * 2 + 0) * 32 + lane) * 4);
        v4f b23_n = *(const v4f*)(sw2 + (size_t)((kn * 2 + 1) * 32 + lane) * 4);
        wmma4(acc2, a, b01, b23);
        a = a_n; b01 = b01_n; b23 = b23_n;
    }

#pragma unroll
    for (int r = 0; r < 8; ++r) {
        int mm = r + (half << 3);
        size_t orow = (size_t)(tile * 16 + mm) * D;
#pragma unroll
        for (int nt = 0; nt < 4; ++nt) {
            h_out[orow + nt * 16 + n0] = acc2[nt][r];  // no lrelu on layer 2
        }
    }
}

extern "C" void kernel_launch(void* const* d_in, const int* in_sizes, int n_in,
                              void* d_out, int out_size, void* d_ws, size_t ws_size,
                              hipStream_t stream) {
    const float* h     = (const float*)d_in[0];
    const float* coord = (const float*)d_in[1];
    const int*   ei    = (const int*)d_in[2];
    const float* w1e   = (const float*)d_in[3];
    const float* b1e   = (const float*)d_in[4];
    const float* w2e   = (const float*)d_in[5];
    const float* b2e   = (const float*)d_in[6];
    const float* w1n   = (const float*)d_in[7];
    const float* b1n   = (const float*)d_in[8];
    const float* w2n   = (const float*)d_in[9];
    const float* b2n   = (const float*)d_in[10];

    float* ws  = (float*)d_ws;
    float* agg = ws + WS_AGG;

    float* h_out     = (float*)d_out;
    float* coord_out = h_out + (size_t)N_NODES * D;
    float* ef_out    = coord_out + (size_t)N_NODES * 3;

    egnn_swizzle_kernel<<<1, 256, 0, stream>>>(w1e, w2e, w1n, w2n, ws);

    const int aggN = N_NODES * D;
    egnn_zero_kernel<<<(aggN + 255) / 256, 256, 0, stream>>>(agg, aggN);

    egnn_edge_kernel<<<N_EDGES / (16 * WPB), BLOCK, 0, stream>>>(
        h, coord, ei, w1e, b1e, b2e, ws + WS_W1E, ws + WS_W2E, agg, ef_out);

    const int ntiles = N_NODES / 16;  // 3125
    egnn_node_kernel<<<(ntiles + WPB - 1) / WPB, BLOCK, 0, stream>>>(
        h, agg, b1n, b2n, ws + WS_W1N, ws + WS_W2N, h_out);

    hipMemcpyAsync(coord_out, coord, (size_t)N_NODES * 3 * sizeof(float),
                   hipMemcpyDeviceToDevice, stream);
}